// DivTree_59416577573287
// MI455X (gfx1250) — compile-verified
//
#include <hip/hip_runtime.h>
#include <hip/hip_bf16.h>

typedef float v2f __attribute__((ext_vector_type(2)));
typedef float v8f __attribute__((ext_vector_type(8)));

#define BATCH    2048
#define N_AGENT  64
#define H_DIM    512
#define N_ACTION 32
#define IN_DIM   (H_DIM + N_AGENT)   // 576
#define BM       32                  // batch rows per block
#define LDS_S    (H_DIM + 4)         // 516-float row stride -> conflict-free b64 reads

static __device__ __forceinline__ v8f wmma_f32(v2f a, v2f b, v8f c) {
    // D = A(16x4, f32) * B(4x16, f32) + C(16x16, f32)
    return __builtin_amdgcn_wmma_f32_16x16x4_f32(
        /*neg_a=*/false, a, /*neg_b=*/false, b,
        /*c_mod=*/(short)0, c, /*reuse_a=*/false, /*reuse_b=*/false);
}

__global__ __launch_bounds__(256) void divtree_mlp_kernel(
    const float* __restrict__ x_in,   // (B, A, H)
    const float* __restrict__ W1,     // (A, IN_DIM, H)
    const float* __restrict__ b1,     // (A, H)
    const float* __restrict__ W2,     // (A, H, NA)
    const float* __restrict__ b2,     // (A, NA)
    const int*   __restrict__ routing,
    const int*   __restrict__ perm,
    float*       __restrict__ out)    // (2, B, A, NA) flat
{
    __shared__ float lds[BM * LDS_S];   // X tile, then reused for H tile (66 KB)

    const int p   = blockIdx.x;          // 0..127 agent-problem
    const int run = p >> 6;              // 0 = x2, 1 = confact
    const int a   = p & 63;              // agent slot
    const int pa  = perm[a];
    const int src_agent = run ? pa : a;  // which agent's x feeds this slot
    const int out_agent = run ? pa : a;  // which agent row receives the result
    const int v   = routing[a];          // weight index

    const int tid   = threadIdx.x;
    const int w     = tid >> 5;          // wave 0..7
    const int lane  = tid & 31;
    const int lhalf = lane >> 4;         // 0: lanes 0-15, 1: lanes 16-31
    const int l16   = lane & 15;

    const int m0 = blockIdx.y * BM;      // first batch row of tile

    // ---- Stage X tile (BM x H) into LDS as float4 ----
    {
        const float4* __restrict__ xg = (const float4*)x_in;
        // global float4 index for [row][col4]: ((b*A + src_agent)*H + col4*4)/4
        for (int it = 0; it < (BM * (H_DIM / 4)) / 256; ++it) {
            int idx  = tid + it * 256;            // 0 .. BM*128-1
            int row  = idx >> 7;                  // /128
            int col4 = idx & 127;
            float4 f = xg[((size_t)(m0 + row) * N_AGENT + src_agent) * (H_DIM / 4) + col4];
            float4* l4 = (float4*)&lds[row * LDS_S];
            l4[col4] = f;
        }
    }
    __syncthreads();

    // ---- Phase A: H = relu(X @ W1v[:512] + b1eff) ----
    // wave w owns output cols [w*64, w*64+64): 4 n-tiles; 2 m-tiles of rows.
    v8f acc[2][4];
    #pragma unroll
    for (int mt = 0; mt < 2; ++mt)
        #pragma unroll
        for (int t = 0; t < 4; ++t)
            acc[mt][t] = (v8f){0.f,0.f,0.f,0.f,0.f,0.f,0.f,0.f};

    const int nbase = w * 64;
    for (int k = 0; k < H_DIM; k += 4) {
        // A fragments: lane -> X[mt*16 + l16][k + 2*lhalf + {0,1}]
        v2f a0 = *(const v2f*)&lds[(0 * 16 + l16) * LDS_S + k + 2 * lhalf];
        v2f a1 = *(const v2f*)&lds[(1 * 16 + l16) * LDS_S + k + 2 * lhalf];
        #pragma unroll
        for (int t = 0; t < 4; ++t) {
            const int n0 = nbase + t * 16;
            const float* bp = W1 + ((size_t)v * IN_DIM + k + 2 * lhalf) * H_DIM + n0 + l16;
            v2f bf;
            bf.x = bp[0];        // B[k + 2*lhalf + 0][n]
            bf.y = bp[H_DIM];    // B[k + 2*lhalf + 1][n]
            acc[0][t] = wmma_f32(a0, bf, acc[0][t]);
            acc[1][t] = wmma_f32(a1, bf, acc[1][t]);
        }
    }

    __syncthreads();   // all waves done reading X from LDS

    // bias (folded one-hot row) + relu, write H tile into LDS
    #pragma unroll
    for (int t = 0; t < 4; ++t) {
        const int n = nbase + t * 16 + l16;
        const float beff = b1[(size_t)v * H_DIM + n]
                         + W1[((size_t)v * IN_DIM + H_DIM + a) * H_DIM + n];
        #pragma unroll
        for (int mt = 0; mt < 2; ++mt) {
            #pragma unroll
            for (int r = 0; r < 8; ++r) {
                float val = acc[mt][t][r] + beff;
                lds[(mt * 16 + r + 8 * lhalf) * LDS_S + n] = fmaxf(val, 0.f);
            }
        }
    }
    __syncthreads();

    // ---- Phase B: out = H @ W2v + b2v  (32x32 tile; waves 0..3) ----
    if (w < 4) {
        const int mt = w >> 1;           // row tile 0/1
        const int nt = w & 1;            // col tile 0/1
        const int o0 = nt * 16;
        v8f c = (v8f){0.f,0.f,0.f,0.f,0.f,0.f,0.f,0.f};
        #pragma unroll 4
        for (int k = 0; k < H_DIM; k += 4) {
            v2f af = *(const v2f*)&lds[(mt * 16 + l16) * LDS_S + k + 2 * lhalf];
            const float* bp = W2 + ((size_t)v * H_DIM + k + 2 * lhalf) * N_ACTION + o0 + l16;
            v2f bf;
            bf.x = bp[0];
            bf.y = bp[N_ACTION];
            c = wmma_f32(af, bf, c);
        }
        const float bias2 = b2[(size_t)v * N_ACTION + o0 + l16];
        float* outp = out + (size_t)run * ((size_t)BATCH * N_AGENT * N_ACTION);
        #pragma unroll
        for (int r = 0; r < 8; ++r) {
            const int brow = m0 + mt * 16 + r + 8 * lhalf;
            outp[((size_t)brow * N_AGENT + out_agent) * N_ACTION + o0 + l16] = c[r] + bias2;
        }
    }
}

extern "C" void kernel_launch(void* const* d_in, const int* in_sizes, int n_in,
                              void* d_out, int out_size, void* d_ws, size_t ws_size,
                              hipStream_t stream) {
    const float* x_in = (const float*)d_in[0];
    const float* W1   = (const float*)d_in[1];
    const float* b1   = (const float*)d_in[2];
    const float* W2   = (const float*)d_in[3];
    const float* b2   = (const float*)d_in[4];
    const int* routing = (const int*)d_in[5];
    const int* perm    = (const int*)d_in[6];
    float* out = (float*)d_out;

    dim3 grid(2 * N_AGENT, BATCH / BM);   // 128 agent-problems x 64 row tiles
    dim3 block(256);
    divtree_mlp_kernel<<<grid, block, 0, stream>>>(x_in, W1, b1, W2, b2, routing, perm, out);
    (void)in_sizes; (void)n_in; (void)out_size; (void)d_ws; (void)ws_size;
}